// CoPE_22625887715427
// MI455X (gfx1250) — compile-verified
//
#include <hip/hip_runtime.h>
#include <hip/hip_bf16.h>

#define CB 2
#define CH 12
#define CS 2048
#define CD 64
#define NPOS 64

typedef float v2f __attribute__((ext_vector_type(2)));
typedef float v8f __attribute__((ext_vector_type(8)));
typedef float f4  __attribute__((ext_vector_type(4)));

// ---------------------------------------------------------------------------
// Kernel 1: logits_int[bh, s, p] = sum_d q[bh, s, d] * P[h, d, p]
// f32 WMMA 16x16x4. One wave computes a 16(M) x 64(N) tile.
// ---------------------------------------------------------------------------
__global__ __launch_bounds__(256) void cope_gemm_wmma(
    const float* __restrict__ q, const float* __restrict__ pemb,
    float* __restrict__ lint)
{
    const int lane = threadIdx.x & 31;
    const int wid  = threadIdx.x >> 5;
    const int tile = blockIdx.x * 8 + wid;     // 16-row tile index
    const int R    = tile << 4;                // global row base (b*h*s flattened)
    const int bh   = R >> 11;                  // R / CS
    const int h    = bh % CH;

    const float* qrow = q + (size_t)R * CD;
    const float* Ph   = pemb + (size_t)h * CD * NPOS;

    const int m  = lane & 15;                  // M (for A) / N (for B) within tile
    const int kk = (lane >> 4) << 1;           // 0 for lanes 0-15, 2 for lanes 16-31
    const int rowAdd = (lane >> 4) << 3;       // upper lane half holds rows M+8 of D

    for (int nt = 0; nt < 4; ++nt) {
        v8f acc = {};
        #pragma unroll
        for (int ks = 0; ks < 16; ++ks) {
            const int k0 = (ks << 2) + kk;
            v2f a, b;
            // A 16x4 f32: VGPR0 = K=kk, VGPR1 = K=kk+1 (per lane half)
            a.x = qrow[(size_t)m * CD + k0];
            a.y = qrow[(size_t)m * CD + k0 + 1];
            // B 4x16 f32: row K striped across lanes, K interleaved by lane half
            b.x = Ph[(size_t)k0 * NPOS + (nt << 4) + m];
            b.y = Ph[(size_t)(k0 + 1) * NPOS + (nt << 4) + m];
            acc = __builtin_amdgcn_wmma_f32_16x16x4_f32(
                false, a, false, b, (short)0, acc, false, false);
        }
        // D 16x16 f32: VGPR r -> row M=r (lanes 0-15), M=r+8 (lanes 16-31)
        float* outBase = lint + (size_t)R * NPOS + (nt << 4) + m;
        #pragma unroll
        for (int rr = 0; rr < 8; ++rr) {
            outBase[(size_t)(rr + rowAdd) * NPOS] = acc[rr];
        }
    }
}

// ---------------------------------------------------------------------------
// Kernel 2: streaming row kernel. One block per (b,h,s) row of 2048 elements.
// sigmoid -> suffix scan -> clip -> gather/lerp from 64-float LDS row -> add.
// ---------------------------------------------------------------------------
__global__ __launch_bounds__(256) void cope_main(
    const float* __restrict__ attn, const float* __restrict__ lint,
    const float* __restrict__ q, const float* __restrict__ pemb,
    float* __restrict__ out, int fused)
{
    __shared__ float Lrow[NPOS];
    __shared__ float waveSums[8];

    const int r    = blockIdx.x;               // row in [0, B*H*S)
    const int t    = threadIdx.x;
    const int lane = t & 31;
    const int wid  = t >> 5;

    // --- stage 0: logits_int row into LDS (precomputed by WMMA, or fused) ---
    if (!fused) {
        if (t < NPOS) Lrow[t] = lint[(size_t)r * NPOS + t];
    } else {
        if (t < NPOS) {
            const int h = (r >> 11) % CH;
            const float* qrow = q + (size_t)r * CD;
            const float* Ph   = pemb + (size_t)h * CD * NPOS + t;
            float acc = 0.f;
            #pragma unroll 8
            for (int d = 0; d < CD; ++d) acc += qrow[d] * Ph[(size_t)d * NPOS];
            Lrow[t] = acc;
        }
    }

    // --- stage 1: load 8 contiguous attn values (streaming, non-temporal) ---
    const float* rowp = attn + (size_t)r * CS + t * 8;
    f4 a0 = __builtin_nontemporal_load((const f4*)rowp);
    f4 a1 = __builtin_nontemporal_load((const f4*)(rowp + 4));

    float g[8];
    #pragma unroll
    for (int i = 0; i < 4; ++i) g[i]     = 1.0f / (1.0f + __expf(-a0[i]));
    #pragma unroll
    for (int i = 0; i < 4; ++i) g[4 + i] = 1.0f / (1.0f + __expf(-a1[i]));

    // --- stage 2: local suffix sums within this thread's 8 elements ---
    float sfx[8];
    sfx[7] = g[7];
    #pragma unroll
    for (int i = 6; i >= 0; --i) sfx[i] = g[i] + sfx[i + 1];
    const float csum = sfx[0];                 // chunk total

    // --- stage 3: block-wide scan of chunk sums (wave32 shuffle + LDS) ---
    float v = csum;
    #pragma unroll
    for (int off = 1; off < 32; off <<= 1) {
        float u = __shfl_up(v, off, 32);
        if (lane >= off) v += u;
    }
    if (lane == 31) waveSums[wid] = v;
    __syncthreads();
    if (t < 32) {
        float w = (lane < 8) ? waveSums[lane] : 0.f;
        #pragma unroll
        for (int off = 1; off < 8; off <<= 1) {
            float u = __shfl_up(w, off, 32);
            if (lane >= off) w += u;
        }
        if (lane < 8) waveSums[lane] = w;
    }
    __syncthreads();
    const float total  = waveSums[7];
    const float prefix = v + ((wid > 0) ? waveSums[wid - 1] : 0.f); // inclusive
    const float tail   = total - prefix;       // sum of chunks strictly after t

    // --- stage 4: clip, gather floor/ceil from LDS, lerp, add, store ---
    f4 o0, o1;
    #pragma unroll
    for (int i = 0; i < 8; ++i) {
        float p  = sfx[i] + tail;              // inclusive suffix sum
        p        = fminf(p, (float)(NPOS - 1));
        float pf = floorf(p);
        int   fi = (int)pf;
        int   ci = (int)ceilf(p);
        float w  = p - pf;
        float val = Lrow[ci] * w + Lrow[fi] * (1.0f - w);
        float res = ((i < 4) ? a0[i] : a1[i - 4]) + val;
        if (i < 4) o0[i] = res; else o1[i - 4] = res;
    }
    float* outp = out + (size_t)r * CS + t * 8;
    __builtin_nontemporal_store(o0, (f4*)outp);
    __builtin_nontemporal_store(o1, (f4*)(outp + 4));
}

// ---------------------------------------------------------------------------
extern "C" void kernel_launch(void* const* d_in, const int* in_sizes, int n_in,
                              void* d_out, int out_size, void* d_ws, size_t ws_size,
                              hipStream_t stream) {
    const float* q    = (const float*)d_in[0];  // [B,H,S,D]
    const float* attn = (const float*)d_in[1];  // [B,H,S,S]
    const float* pemb = (const float*)d_in[2];  // [1,H,D,NPOS]
    float* out = (float*)d_out;                 // [B,H,S,S]
    float* lint = (float*)d_ws;                 // [B,H,S,NPOS] scratch

    const size_t need = (size_t)CB * CH * CS * NPOS * sizeof(float);
    const int fused = (ws_size < need) ? 1 : 0;

    if (!fused) {
        // (B*H*S/16) 16-row tiles, 8 waves (tiles) per 256-thread block
        const int tiles  = (CB * CH * CS) / 16;   // 3072
        const int blocks = tiles / 8;             // 384
        cope_gemm_wmma<<<blocks, 256, 0, stream>>>(q, pemb, lint);
    }
    cope_main<<<CB * CH * CS, 256, 0, stream>>>(attn, lint, q, pemb, out, fused);
}